// GATWithJK_88132728914599
// MI455X (gfx1250) — compile-verified
//
#include <hip/hip_runtime.h>
#include <hip/hip_bf16.h>
#include <math.h>

// ---------------------------------------------------------------------------
// GAT-with-JK forward for MI455X (gfx1250).
// GEMMs: V_WMMA_F32_16X16X4_F32, guard-free inner loop, register-blocked
// 16x64 output strip per wave (1 A-frag load feeds 4 WMMAs per K-step).
// ---------------------------------------------------------------------------

typedef __attribute__((ext_vector_type(2))) float v2f;
typedef __attribute__((ext_vector_type(8))) float v8f;

#define N_NODES   50000
#define N_EDGES   800000
#define E_TOT     (N_EDGES + N_NODES)   // + self loops
#define NUM_GRAPHS 512
#define HID       64
#define HEADS     4
#define HH        (HID * HEADS)         // 256
#define FCDIM     768
#define NEG_SLOPE 0.2f
#define LDH0      20                    // h0 padded from 18 -> 20 (K % 4 == 0)

// ---------------- helpers ----------------

__device__ __forceinline__ float atomic_max_f32(float* addr, float val) {
  // sign-split trick: valid total order incl. negatives / -FLT_MAX init
  if (val >= 0.0f)
    return __int_as_float(atomicMax((int*)addr, __float_as_int(val)));
  else
    return __uint_as_float(atomicMin((unsigned int*)addr, __float_as_uint(val)));
}

__device__ __forceinline__ void edge_ends(const int* __restrict__ ei, int eid,
                                          int* s, int* d) {
  if (eid < N_EDGES) { *s = ei[eid]; *d = ei[N_EDGES + eid]; }
  else               { *s = *d = eid - N_EDGES; }   // self loop
}

__device__ __forceinline__ float leaky(float v) {
  return v > 0.0f ? v : NEG_SLOPE * v;
}

// ---------------- fill ----------------

__global__ void k_fill(float* __restrict__ p, float v, int n) {
  for (int i = blockIdx.x * blockDim.x + threadIdx.x; i < n;
       i += gridDim.x * blockDim.x)
    p[i] = v;
}

// ---------------- embedding concat (padded to LDH0 cols) ----------------

__global__ void k_embed(const float* __restrict__ x, const float* __restrict__ emb,
                        float* __restrict__ h0, int n) {
  int i = blockIdx.x * blockDim.x + threadIdx.x;
  if (i >= n) return;
  int id = (int)x[i * 11];
  float* o = h0 + i * LDH0;
  const float* e = emb + id * 8;
#pragma unroll
  for (int c = 0; c < 8; ++c) o[c] = e[c];
#pragma unroll
  for (int c = 0; c < 10; ++c) o[8 + c] = x[i * 11 + 1 + c];
  o[18] = 0.0f;
  o[19] = 0.0f;
}

// ---------------- weight transpose + zero-pad: Bt[Np,Kp] = B[K,N]^T ----------

__global__ void k_tpad(const float* __restrict__ B, float* __restrict__ Bt,
                       int K, int N, int Kp, int Np) {
  int i = blockIdx.x * blockDim.x + threadIdx.x;
  if (i >= Np * Kp) return;
  int n = i / Kp, k = i - n * Kp;
  Bt[i] = (k < K && n < N) ? B[k * N + n] : 0.0f;
}

// ---------------- WMMA fp32 GEMM (register-blocked over N) -------------------
// C[M, 16*NB*tilesNB] = A[M,K] @ Bt^T (+bias)(+relu).  Bt is [N,K] transposed,
// K%4==0.  One wave computes a 16 x (16*NB) strip: per K-step it loads one A
// fragment (b64) and NB B fragments (b64), issuing NB v_wmma_f32_16x16x4_f32.
// Frags per ISA 7.12.2: lane&15 = M row (A) / N col (B); lane half selects
// K {+0,+1} vs {+2,+3}; C/D reg i -> M = i + 8*(lane>=16), N = lane&15.

template <int NB>
__global__ void k_gemm_wmma(const float* __restrict__ A, int lda,
                            const float* __restrict__ Bt, int K,
                            const float* __restrict__ bias,
                            float* __restrict__ C, int ldc,
                            int tilesNB, int totalJobs, int relu) {
  const int lane  = threadIdx.x & 31;
  const int wave  = threadIdx.x >> 5;
  const int gwave = blockIdx.x * (blockDim.x >> 5) + wave;
  if (gwave >= totalJobs) return;         // wave-uniform: EXEC all-1s at WMMA
  const int tm = (gwave / tilesNB) << 4;
  const int tn = (gwave % tilesNB) * (16 * NB);

  const int half = lane >> 4;             // 0: K+0/K+1   1: K+2/K+3
  const int l    = lane & 15;
  const float* Ap = A  + (long)(tm + l) * lda + half * 2;
  const float* Bp = Bt + (long)(tn + l) * K   + half * 2;

  v8f acc[NB];
#pragma unroll
  for (int j = 0; j < NB; ++j) acc[j] = (v8f){};

#pragma unroll 2
  for (int k = 0; k < K; k += 4) {
    v2f a = *(const v2f*)(Ap + k);
#pragma unroll
    for (int j = 0; j < NB; ++j) {
      v2f b = *(const v2f*)(Bp + (long)j * 16 * K + k);
      acc[j] = __builtin_amdgcn_wmma_f32_16x16x4_f32(
          /*neg_a=*/false, a, /*neg_b=*/false, b,
          /*c_mod=*/(short)0, acc[j], /*reuse_a=*/false, /*reuse_b=*/false);
    }
  }

#pragma unroll
  for (int j = 0; j < NB; ++j) {
    const int col = tn + j * 16 + l;
    float bv = bias ? bias[col] : 0.0f;
#pragma unroll
    for (int i = 0; i < 8; ++i) {
      int row = tm + i + half * 8;
      float v = acc[j][i] + bv;
      if (relu) v = fmaxf(v, 0.0f);
      C[(long)row * ldc + col] = v;
    }
  }
}

// ---------------- attention coefficients a_s, a_d ----------------

__global__ void k_attn(const float* __restrict__ h, const float* __restrict__ a_src,
                       const float* __restrict__ a_dst,
                       float* __restrict__ as_, float* __restrict__ ad_, int n) {
  int i = blockIdx.x * blockDim.x + threadIdx.x;
  if (i >= n * HEADS) return;
  int node = i >> 2, hd = i & 3;
  const float* hp = h + node * HH + hd * HID;
  const float* s = a_src + hd * HID;
  const float* d = a_dst + hd * HID;
  float ss = 0.0f, dd = 0.0f;
#pragma unroll 8
  for (int c = 0; c < HID; ++c) { float v = hp[c]; ss += v * s[c]; dd += v * d[c]; }
  as_[i] = ss;
  ad_[i] = dd;
}

// ---------------- edge softmax (3 passes, logits recomputed) ----------------

__global__ void k_edge_max(const int* __restrict__ ei,
                           const float* __restrict__ as_, const float* __restrict__ ad_,
                           float* __restrict__ m) {
  int i = blockIdx.x * blockDim.x + threadIdx.x;
  if (i >= E_TOT * HEADS) return;
  int eid = i >> 2, hd = i & 3;
  int s, d; edge_ends(ei, eid, &s, &d);
  float e = leaky(as_[s * HEADS + hd] + ad_[d * HEADS + hd]);
  atomic_max_f32(&m[d * HEADS + hd], e);
}

__global__ void k_edge_denom(const int* __restrict__ ei,
                             const float* __restrict__ as_, const float* __restrict__ ad_,
                             const float* __restrict__ m, float* __restrict__ denom) {
  int i = blockIdx.x * blockDim.x + threadIdx.x;
  if (i >= E_TOT * HEADS) return;
  int eid = i >> 2, hd = i & 3;
  int s, d; edge_ends(ei, eid, &s, &d);
  float e = leaky(as_[s * HEADS + hd] + ad_[d * HEADS + hd]);
  atomicAdd(&denom[d * HEADS + hd], expf(e - m[d * HEADS + hd]));
}

__global__ void k_edge_msg(const int* __restrict__ ei,
                           const float* __restrict__ as_, const float* __restrict__ ad_,
                           const float* __restrict__ m, const float* __restrict__ denom,
                           const float* __restrict__ h, float* __restrict__ agg) {
  int i = blockIdx.x * blockDim.x + threadIdx.x;
  if (i >= E_TOT * HEADS) return;
  int eid = i >> 2, hd = i & 3;
  int s, d; edge_ends(ei, eid, &s, &d);
  float e = leaky(as_[s * HEADS + hd] + ad_[d * HEADS + hd]);
  float alpha = expf(e - m[d * HEADS + hd]) / (denom[d * HEADS + hd] + 1e-16f);
  const float* hp = h + s * HH + hd * HID;
  float* op = agg + d * HH + hd * HID;
#pragma unroll 4
  for (int c = 0; c < HID; ++c)
    atomicAdd(&op[c], hp[c] * alpha);
}

// ---------------- bias + relu into JK concat slice ----------------

__global__ void k_bias_relu(const float* __restrict__ agg, const float* __restrict__ bias,
                            float* __restrict__ cat, int off, int n) {
  int i = blockIdx.x * blockDim.x + threadIdx.x;
  if (i >= n * HH) return;
  int node = i >> 8, c = i & 255;
  cat[node * FCDIM + off + c] = fmaxf(agg[i] + bias[c], 0.0f);
}

// ---------------- global mean pool ----------------

__global__ void k_pool_sum(const float* __restrict__ cat, const int* __restrict__ batch,
                           float* __restrict__ sums, float* __restrict__ cnt, int n) {
  int i = blockIdx.x * blockDim.x + threadIdx.x;
  if (i >= n * FCDIM) return;
  int node = i / FCDIM, c = i - node * FCDIM;
  int b = batch[node];
  atomicAdd(&sums[b * FCDIM + c], cat[i]);
  if (c == 0) atomicAdd(&cnt[b], 1.0f);
}

__global__ void k_pool_div(const float* __restrict__ sums, const float* __restrict__ cnt,
                           float* __restrict__ g, int total) {
  int i = blockIdx.x * blockDim.x + threadIdx.x;
  if (i >= total) return;
  int b = i / FCDIM;
  g[i] = sums[i] / fmaxf(cnt[b], 1.0f);
}

// ---------------- final 2-column extract + bias ----------------

__global__ void k_out2(const float* __restrict__ gtmp, const float* __restrict__ bias,
                       float* __restrict__ out) {
  int i = blockIdx.x * blockDim.x + threadIdx.x;
  if (i >= NUM_GRAPHS * 2) return;
  int g = i >> 1, c = i & 1;
  out[i] = gtmp[g * 16 + c] + bias[c];
}

// ---------------------------------------------------------------------------

extern "C" void kernel_launch(void* const* d_in, const int* in_sizes, int n_in,
                              void* d_out, int out_size, void* d_ws, size_t ws_size,
                              hipStream_t stream) {
  // inputs (setup_inputs dict order, params flattened depth-first)
  const float* x     = (const float*)d_in[0];   // [50000, 11]
  const int*   ei    = (const int*)  d_in[1];   // [2, 800000]
  const int*   batch = (const int*)  d_in[2];   // [50000]
  const float* emb   = (const float*)d_in[3];   // [2048, 8]
  const float* cW[3]  = {(const float*)d_in[4],  (const float*)d_in[8],  (const float*)d_in[12]};
  const float* cAs[3] = {(const float*)d_in[5],  (const float*)d_in[9],  (const float*)d_in[13]};
  const float* cAd[3] = {(const float*)d_in[6],  (const float*)d_in[10], (const float*)d_in[14]};
  const float* cB[3]  = {(const float*)d_in[7],  (const float*)d_in[11], (const float*)d_in[15]};
  const float* fW[3]  = {(const float*)d_in[16], (const float*)d_in[18], (const float*)d_in[20]};
  const float* fB[3]  = {(const float*)d_in[17], (const float*)d_in[19], (const float*)d_in[21]};
  float* out = (float*)d_out;                   // [512, 2]

  // workspace layout (floats; all offsets 8B-aligned)
  float* ws   = (float*)d_ws;
  float* h0   = ws;                        // 50000*20
  float* hbuf = h0   + N_NODES * LDH0;     // 50000*256
  float* agg  = hbuf + N_NODES * HH;       // 50000*256
  float* cat  = agg  + N_NODES * HH;       // 50000*768
  float* as_  = cat  + N_NODES * FCDIM;    // 50000*4
  float* ad_  = as_  + N_NODES * HEADS;
  float* mbuf = ad_  + N_NODES * HEADS;
  float* den  = mbuf + N_NODES * HEADS;
  float* psum = den  + N_NODES * HEADS;    // 512*768
  float* pcnt = psum + NUM_GRAPHS * FCDIM; // 512
  float* g0   = pcnt + NUM_GRAPHS;         // 512*768
  float* g1   = g0   + NUM_GRAPHS * FCDIM; // 512*768
  float* wtr  = g1   + NUM_GRAPHS * FCDIM; // 768*768 (transposed-weight scratch)
  float* gtmp = wtr  + FCDIM * FCDIM;      // 512*16

  const int BT = 256;
  auto blocks = [](long n, int bt) { return (int)((n + bt - 1) / bt); };

  // 1) embedding concat (padded)
  k_embed<<<blocks(N_NODES, BT), BT, 0, stream>>>(x, emb, h0, N_NODES);

  // 2) GAT layers
  const int EH = E_TOT * HEADS;
  for (int l = 0; l < 3; ++l) {
    const float* Ain = (l == 0) ? h0 : (cat + (l - 1) * HH);
    const int    lda = (l == 0) ? LDH0 : FCDIM;
    const int    K   = (l == 0) ? 18 : HH;
    const int    Kp  = (l == 0) ? LDH0 : HH;

    // transpose + pad layer weight: wtr[256, Kp]
    k_tpad<<<blocks((long)HH * Kp, BT), BT, 0, stream>>>(cW[l], wtr, K, HH, Kp, HH);

    // h = Ain @ W  (wave computes 16x64 strip; 1 A-load : 4 WMMA per K-step)
    const int jobs = (N_NODES / 16) * (HH / 64);   // 3125 * 4 = 12500
    k_gemm_wmma<4><<<blocks((long)jobs, 8), 8 * 32, 0, stream>>>(
        Ain, lda, wtr, Kp, nullptr, hbuf, HH, HH / 64, jobs, 0);

    // attention coefficients
    k_attn<<<blocks((long)N_NODES * HEADS, BT), BT, 0, stream>>>(
        hbuf, cAs[l], cAd[l], as_, ad_, N_NODES);

    // init segment-softmax state + aggregation buffer
    k_fill<<<blocks((long)N_NODES * HEADS, BT), BT, 0, stream>>>(mbuf, -3.402823466e38f, N_NODES * HEADS);
    k_fill<<<blocks((long)N_NODES * HEADS, BT), BT, 0, stream>>>(den, 0.0f, N_NODES * HEADS);
    k_fill<<<blocks((long)N_NODES * HH, BT), BT, 0, stream>>>(agg, 0.0f, N_NODES * HH);

    // edge softmax + weighted aggregation
    k_edge_max  <<<blocks(EH, BT), BT, 0, stream>>>(ei, as_, ad_, mbuf);
    k_edge_denom<<<blocks(EH, BT), BT, 0, stream>>>(ei, as_, ad_, mbuf, den);
    k_edge_msg  <<<blocks(EH, BT), BT, 0, stream>>>(ei, as_, ad_, mbuf, den, hbuf, agg);

    // bias + relu -> JK concat slice
    k_bias_relu<<<blocks((long)N_NODES * HH, BT), BT, 0, stream>>>(
        agg, cB[l], cat, l * HH, N_NODES);
  }

  // 3) global mean pool
  k_fill<<<blocks((long)NUM_GRAPHS * FCDIM, BT), BT, 0, stream>>>(psum, 0.0f, NUM_GRAPHS * FCDIM);
  k_fill<<<blocks(NUM_GRAPHS, BT), BT, 0, stream>>>(pcnt, 0.0f, NUM_GRAPHS);
  k_pool_sum<<<blocks((long)N_NODES * FCDIM, BT), BT, 0, stream>>>(cat, batch, psum, pcnt, N_NODES);
  k_pool_div<<<blocks((long)NUM_GRAPHS * FCDIM, BT), BT, 0, stream>>>(psum, pcnt, g0, NUM_GRAPHS * FCDIM);

  // 4) FC head (WMMA)
  {
    const int jobs = (NUM_GRAPHS / 16) * (FCDIM / 64);   // 32 * 12 = 384
    k_tpad<<<blocks((long)FCDIM * FCDIM, BT), BT, 0, stream>>>(fW[0], wtr, FCDIM, FCDIM, FCDIM, FCDIM);
    k_gemm_wmma<4><<<blocks((long)jobs, 8), 8 * 32, 0, stream>>>(
        g0, FCDIM, wtr, FCDIM, fB[0], g1, FCDIM, FCDIM / 64, jobs, 1);
    k_tpad<<<blocks((long)FCDIM * FCDIM, BT), BT, 0, stream>>>(fW[1], wtr, FCDIM, FCDIM, FCDIM, FCDIM);
    k_gemm_wmma<4><<<blocks((long)jobs, 8), 8 * 32, 0, stream>>>(
        g1, FCDIM, wtr, FCDIM, fB[1], g0, FCDIM, FCDIM / 64, jobs, 1);

    // final 768 -> 2: pad N to 16 with zero columns, then extract
    const int jobs2 = NUM_GRAPHS / 16;                   // 32
    k_tpad<<<blocks((long)16 * FCDIM, BT), BT, 0, stream>>>(fW[2], wtr, FCDIM, 2, FCDIM, 16);
    k_gemm_wmma<1><<<blocks((long)jobs2, 8), 8 * 32, 0, stream>>>(
        g0, FCDIM, wtr, FCDIM, nullptr, gtmp, 16, 1, jobs2, 0);
    k_out2<<<blocks((long)NUM_GRAPHS * 2, BT), BT, 0, stream>>>(gtmp, fB[2], out);
  }
}